// AndLayer_7249904796446
// MI455X (gfx1250) — compile-verified
//
#include <hip/hip_runtime.h>
#include <hip/hip_bf16.h>

// ---------------------------------------------------------------------------
// AndLayer fused kernel for gfx1250 (MI455X).
//
//   out[b][c] = max_{k<56} min_{i<448} ( X[b][k][i] * tanh(W[i][c])
//                                        + 1 - tanh(W[i][c])^2 )
//
// X[b][k] is a permutation-gather of nullary/unary/binary predicates.
// Tropical (min/max) reduction => WMMA not applicable; we use the CDNA5
// async data path instead: per-lane global_load_async_to_lds_b128 gathers
// the 100KB X tile straight into LDS (ASYNCcnt tracked), gfx1250 hardware
// v_tanh_f32, and ds_load_b128 broadcast reads in the hot loop.
// ---------------------------------------------------------------------------

#define N_OBJ   8
#define P0      64
#define P1      128
#define P2      64
#define CCH     128               // output channels
#define KPERM   56                // 8*7 ordered pairs
#define NIN     448               // P0 + 2*P1 + 2*P2
#define KSPLIT  4                 // k-groups per block
#define KPG     (KPERM / KSPLIT)  // 14 accumulators / thread
#define THREADS (CCH * KSPLIT)    // 512 threads = 16 wave32
#define QPR     (NIN / 4)         // 112 b128 transfers per k-row
#define NXFER   (KPERM * QPR)     // 6272 b128 transfers per block

__device__ __forceinline__ float fast_tanh(float x) {
#if __has_builtin(__builtin_amdgcn_tanhf)
  return __builtin_amdgcn_tanhf(x);        // gfx1250 v_tanh_f32
#elif __has_builtin(__builtin_amdgcn_tanh_f32)
  return __builtin_amdgcn_tanh_f32(x);
#else
  return tanhf(x);
#endif
}

__global__ __launch_bounds__(THREADS) void and_layer_kernel(
    const float* __restrict__ nullary,   // (B, 64)
    const float* __restrict__ unary,     // (B, 8, 128)
    const float* __restrict__ binary,    // (B, 8, 7, 64) == (B, 56, 64)
    const float* __restrict__ kern,      // (448, 128)
    float* __restrict__ out) {           // (B, 128)
  extern __shared__ float smem[];        // X[56][448] | red[4][128]
  const int b   = blockIdx.x;
  const int tid = threadIdx.x;

  // LDS byte offset of smem[0]: generic AS3->flat pointers carry the LDS
  // offset in the low 32 bits on AMDGPU.
  const unsigned lds_base = (unsigned)(unsigned long long)smem;

  // Warm the tanh-weight matrix into cache while the async gather runs.
  for (int p = tid * 64; p < NIN * CCH; p += THREADS * 64)
    __builtin_prefetch(kern + p, 0, 1);  // -> global_prefetch_b8

  // ---- Stage X[b] into LDS with per-lane async b128 gathers --------------
  // Transfer j covers X[k][4q .. 4q+3].  Segment map over q (i = 4q):
  //   q<16  : nullary[b][4q]                     (i in [0,64))
  //   q<48  : unary[b][o0][4(q-16)]              (i in [64,192))
  //   q<80  : unary[b][o1][4(q-48)]              (i in [192,320))
  //   q<96  : binary[b][k ][4(q-80)]             (i in [320,384))  row = k
  //   else  : binary[b][k2][4(q-96)]             (i in [384,448))  row = rev-perm
  for (int j = tid; j < NXFER; j += THREADS) {
    const int k  = j / QPR;
    const int q  = j - k * QPR;
    const int o0 = k / 7;
    const int rr = k - o0 * 7;
    const int o1 = rr + (rr >= o0 ? 1 : 0);

    const float* gp;
    if (q < 16) {
      gp = nullary + b * P0 + q * 4;
    } else if (q < 48) {
      gp = unary + (b * N_OBJ + o0) * P1 + (q - 16) * 4;
    } else if (q < 80) {
      gp = unary + (b * N_OBJ + o1) * P1 + (q - 48) * 4;
    } else if (q < 96) {
      gp = binary + (b * KPERM + k) * P2 + (q - 80) * 4;
    } else {
      const int k2 = o1 * 7 + o0 - (o0 > o1 ? 1 : 0);   // index of (o1,o0)
      gp = binary + (b * KPERM + k2) * P2 + (q - 96) * 4;
    }

    const unsigned           ldsa = lds_base + (unsigned)(k * NIN + q * 4) * 4u;
    const unsigned long long ga   = (unsigned long long)gp;
    asm volatile("global_load_async_to_lds_b128 %0, %1, off"
                 :: "v"(ldsa), "v"(ga)
                 : "memory");
  }
  asm volatile("s_wait_asynccnt 0x0" ::: "memory");
  __syncthreads();

  // ---- Fused tanh + weighted-truth + min_i, 14 k's per thread ------------
  const int c  = tid & (CCH - 1);
  const int kg = tid >> 7;         // 0..3, wave-uniform (wave32 | 128)
  const int k0 = kg * KPG;

  float m[KPG];
#pragma unroll
  for (int kk = 0; kk < KPG; ++kk) m[kk] = __builtin_inff();

  const float* __restrict__ kcol = kern + c;       // column c, stride 128
  const float* __restrict__ Xrow = smem + k0 * NIN;

  // 4 i's per iteration: one broadcast ds_load_b128 per k-row, 4 v_tanh
  // (TRANS, co-executes with the fma/min VALU stream).
  for (int i = 0; i < NIN; i += 4) {
    const float w0 = fast_tanh(kcol[(i + 0) * CCH]);
    const float w1 = fast_tanh(kcol[(i + 1) * CCH]);
    const float w2 = fast_tanh(kcol[(i + 2) * CCH]);
    const float w3 = fast_tanh(kcol[(i + 3) * CCH]);
    const float b0 = 1.0f - w0 * w0;
    const float b1 = 1.0f - w1 * w1;
    const float b2 = 1.0f - w2 * w2;
    const float b3 = 1.0f - w3 * w3;
#pragma unroll
    for (int kk = 0; kk < KPG; ++kk) {
      // 16B-aligned, wave-uniform LDS address -> single broadcast b128 read.
      const float4 x = *(const float4*)&Xrow[kk * NIN + i];
      float mv = m[kk];
      mv = fminf(mv, fmaf(x.x, w0, b0));
      mv = fminf(mv, fmaf(x.y, w1, b1));
      mv = fminf(mv, fmaf(x.z, w2, b2));
      mv = fminf(mv, fmaf(x.w, w3, b3));
      m[kk] = mv;
    }
  }

  float acc = m[0];
#pragma unroll
  for (int kk = 1; kk < KPG; ++kk) acc = fmaxf(acc, m[kk]);

  // ---- max over the 4 k-groups via LDS -----------------------------------
  float* red = smem + KPERM * NIN;
  red[kg * CCH + c] = acc;
  __syncthreads();
  if (kg == 0) {
    float v = red[c];
    v = fmaxf(v, red[CCH + c]);
    v = fmaxf(v, red[2 * CCH + c]);
    v = fmaxf(v, red[3 * CCH + c]);
    out[b * CCH + c] = v;
  }
}

extern "C" void kernel_launch(void* const* d_in, const int* in_sizes, int n_in,
                              void* d_out, int out_size, void* d_ws, size_t ws_size,
                              hipStream_t stream) {
  (void)n_in; (void)out_size; (void)d_ws; (void)ws_size;
  const float* nullary = (const float*)d_in[0];
  const float* unary   = (const float*)d_in[1];
  const float* binary  = (const float*)d_in[2];
  const float* kern    = (const float*)d_in[3];
  float* out = (float*)d_out;

  const int B = in_sizes[0] / P0;  // 32 for the reference shapes
  const size_t shmem = (size_t)(KPERM * NIN + KSPLIT * CCH) * sizeof(float); // 102400 B

  and_layer_kernel<<<B, THREADS, shmem, stream>>>(nullary, unary, binary, kern, out);
}